// ClusterSeedClsPlus_32555852103943
// MI455X (gfx1250) — compile-verified
//
#include <hip/hip_runtime.h>
#include <hip/hip_bf16.h>

// ---------------- problem constants (from reference) ----------------
#define HH 1024
#define WW 3072
#define NPIX (HH * WW)          // 3145728
#define NWORDS (NPIX / 32)      // 98304 bitmap words
#define MAX_ITER 4096
#define MIN_PIXEL 160u
#define MIN_INST_PIXEL 160u
#define THRESHOLD 0.5f
#define MIN_SEED_THRESH 0.5f
#define INST_RATIO 0.5f
#define LN2F 0.6931471805599453f   // dist > 0.5  <=>  quad < ln 2
#define ABLOCKS 1024
#define CBLOCKS 1024
#define CCHUNK 2048                // pixels staged to LDS per block-step

typedef unsigned int u32;
typedef unsigned int u32x4 __attribute__((ext_vector_type(4)));
typedef int i32x8 __attribute__((ext_vector_type(8)));
typedef int i32x4 __attribute__((ext_vector_type(4)));

struct Scal {
    u32 uncl_count;     // unclustered.sum()
    u32 done;
    u32 active;         // loop condition result for this iteration
    int count;          // next instance id
    int count_snapshot; // id used for assignment this iteration
    float cx, cy, sx, sy;
    u32 stop;
    u32 psum, usum;
    u32 sidx;
};

// ---------------- CDNA5-specific helpers ----------------
__device__ __forceinline__ u32 ballot32(bool p) {
#if __has_builtin(__builtin_amdgcn_ballot_w32)
    return __builtin_amdgcn_ballot_w32(p);
#else
    return (u32)__ballot(p);   // wave32: low 32 bits
#endif
}

// Async copy 16B global -> LDS (GLOBAL_LOAD_ASYNC_TO_LDS_B128, ASYNCcnt).
__device__ __forceinline__ void async_b128(u32 lds_byte_addr, const void* gaddr) {
    asm volatile("global_load_async_to_lds_b128 %0, %1, off"
                 :: "v"(lds_byte_addr), "v"(gaddr) : "memory");
}

__device__ __forceinline__ void wait_async0() {
#if __has_builtin(__builtin_amdgcn_s_wait_asynccnt)
    __builtin_amdgcn_s_wait_asynccnt(0);
#else
    asm volatile("s_wait_asynccnt 0" ::: "memory");
#endif
}

__device__ __forceinline__ void wait_tensor0() {
#if __has_builtin(__builtin_amdgcn_s_wait_tensorcnt)
    __builtin_amdgcn_s_wait_tensorcnt(0);
#else
    asm volatile("s_wait_tensorcnt 0" ::: "memory");
#endif
}

#if __has_builtin(__builtin_amdgcn_tensor_load_to_lds)
#define HAS_TDM 1
#else
#define HAS_TDM 0
#endif

#if HAS_TDM
// One-descriptor TDM DMA of a 1-D tile of `nelem` f32 from global to LDS.
// D# layout per CDNA5 ISA ch.8: group0 = {flags, lds_addr, global_addr, type=2},
// group1 = {data_size=4B, tensor_dim0=tile_dim0=nelem, stride=nelem}, groups 2/3 zero.
__device__ __forceinline__ void tdm_load_1d(u32 lds_addr, const void* gsrc, u32 nelem) {
    unsigned long long ga = (unsigned long long)(uintptr_t)gsrc;
    u32x4 g0;
    g0.x = 1u;                                            // count=1, user mode, no gather
    g0.y = lds_addr;                                      // lds byte address
    g0.z = (u32)ga;                                       // global_addr[31:0]
    g0.w = ((u32)(ga >> 32) & 0x01FFFFFFu) | (2u << 30);  // global_addr[56:32] | type=2
    i32x8 g1;
    g1[0] = (int)(2u << 16);                              // data_size code 2 = 4 bytes
    g1[1] = (int)((nelem & 0xFFFFu) << 16);               // tensor_dim0[15:0]
    g1[2] = (int)(((nelem >> 16) & 0xFFFFu) | (1u << 16)); // dim0[31:16] | tensor_dim1=1
    g1[3] = (int)((nelem & 0xFFFFu) << 16);               // tile_dim0 = nelem (fits 16b)
    g1[4] = 0;                                            // tile_dim1/2 unused (1-D tile)
    g1[5] = (int)nelem;                                   // tensor_dim0_stride[31:0]
    g1[6] = 0;
    g1[7] = 0;
    i32x4 z4 = {0, 0, 0, 0};
#if defined(__clang_major__) && (__clang_major__ >= 23)
    i32x8 z8 = {0, 0, 0, 0, 0, 0, 0, 0};
    __builtin_amdgcn_tensor_load_to_lds(g0, g1, z4, z4, z8, 0);
#else
    __builtin_amdgcn_tensor_load_to_lds(g0, g1, z4, z4, 0);
#endif
    wait_tensor0();                                       // TENSORcnt drain (issuing wave)
}
#endif

// ---------------- kernels ----------------
__global__ void k_init(Scal* sc) {
    sc->uncl_count = 0u; sc->done = 0u; sc->active = 1u;
    sc->count = 1; sc->count_snapshot = 1;
    sc->cx = 0.f; sc->cy = 0.f; sc->sx = 0.f; sc->sy = 0.f;
    sc->stop = 0u; sc->psum = 0u; sc->usum = 0u; sc->sidx = 0u;
}

// emb = tanh(pred[0:2]) + grid ; seed = sigmoid(pred[6]) ; m = seed > 0.5
__global__ void k_prep(const float* __restrict__ pred,
                       float* __restrict__ embx, float* __restrict__ emby,
                       float* __restrict__ seed, int* __restrict__ inst,
                       u32* __restrict__ m_bits, u32* __restrict__ uncl_bits,
                       Scal* sc) {
    int i = blockIdx.x * 256 + threadIdx.x;   // grid covers NPIX exactly
    float p0 = pred[i];
    float p1 = pred[(size_t)NPIX + i];
    float p6 = pred[(size_t)6 * NPIX + i];
    int x = i % WW, y = i / WW;
    embx[i] = tanhf(p0) + (float)x * (3.0f / 3071.0f);  // linspace(0,3,3072)
    emby[i] = tanhf(p1) + (float)y * (1.0f / 1023.0f);  // linspace(0,1,1024)
    float sd = 1.0f / (1.0f + expf(-p6));
    seed[i] = sd;
    inst[i] = 0;
    bool mm = sd > MIN_SEED_THRESH;
    u32 bal = ballot32(mm);                   // wave32 bitmap pack
    if ((threadIdx.x & 31) == 0) {
        m_bits[i >> 5] = bal;
        uncl_bits[i >> 5] = bal;
        atomicAdd(&sc->uncl_count, (u32)__popc(bal));
    }
}

// per-block partial argmax of seed * unclustered
__global__ void k_argmax(const Scal* sc,
                         const float* __restrict__ seed,
                         const u32* __restrict__ uncl_bits,
                         float* __restrict__ partv, u32* __restrict__ parti) {
    if (sc->done || sc->uncl_count <= MIN_PIXEL) return;
    __shared__ float sv[256];
    __shared__ u32 si[256];
    int tid = threadIdx.x;
    float bv = -1.0f; u32 bi = 0u;
    const int stride = ABLOCKS * 256;
    for (int i = blockIdx.x * 256 + tid; i < NPIX; i += stride) {
        if (i + stride < NPIX) __builtin_prefetch(seed + i + stride, 0, 0); // global_prefetch_b8
        u32 uw = uncl_bits[(u32)i >> 5];
        if ((uw >> (i & 31)) & 1u) {
            float v = seed[i];
            if (v > bv) { bv = v; bi = (u32)i; }   // ascending i -> first max
        }
    }
    sv[tid] = bv; si[tid] = bi; __syncthreads();
    for (int off = 128; off > 0; off >>= 1) {
        if (tid < off) {
            float v = sv[tid + off]; u32 ix = si[tid + off];
            if (v > sv[tid] || (v == sv[tid] && ix < si[tid])) { sv[tid] = v; si[tid] = ix; }
        }
        __syncthreads();
    }
    if (tid == 0) { partv[blockIdx.x] = sv[0]; parti[blockIdx.x] = si[0]; }
}

// single block: loop condition, finalize argmax, pick center/sigma, clear seed pixel
__global__ void k_select(Scal* sc, const float* __restrict__ pred,
                         const float* __restrict__ embx, const float* __restrict__ emby,
                         const float* __restrict__ partv, const u32* __restrict__ parti,
                         u32* __restrict__ uncl_bits) {
    __shared__ float sv[256];
    __shared__ u32 si[256];
    __shared__ int s_act;
    int tid = threadIdx.x;
    if (tid == 0) {
        int act = (sc->done == 0u) && (sc->uncl_count > MIN_PIXEL);
        sc->active = (u32)act; s_act = act;
    }
    __syncthreads();
    if (!s_act) return;
    float bv = -1.0f; u32 bi = 0u;
    for (int k = tid; k < ABLOCKS; k += 256) {
        float v = partv[k]; u32 ix = parti[k];
        if (v > bv || (v == bv && ix < bi)) { bv = v; bi = ix; }
    }
    sv[tid] = bv; si[tid] = bi; __syncthreads();
    for (int off = 128; off > 0; off >>= 1) {
        if (tid < off) {
            float v = sv[tid + off]; u32 ix = si[tid + off];
            if (v > sv[tid] || (v == sv[tid] && ix < si[tid])) { sv[tid] = v; si[tid] = ix; }
        }
        __syncthreads();
    }
    if (tid == 0) {
        float score = sv[0]; u32 sidx = si[0];
        if (score < THRESHOLD) {
            sc->stop = 1u; sc->done = 1u;           // uncl reverts => leave untouched
        } else {
            sc->stop = 0u;
            uncl_bits[sidx >> 5] &= ~(1u << (sidx & 31));   // uncl1: clear seed pixel
            sc->uncl_count -= 1u;
            sc->cx = embx[sidx]; sc->cy = emby[sidx];
            sc->sx = expf(10.0f * pred[(size_t)2 * NPIX + sidx]);
            sc->sy = expf(10.0f * pred[(size_t)3 * NPIX + sidx]);
            sc->psum = 0u; sc->usum = 0u;
            sc->count_snapshot = sc->count;
            sc->sidx = sidx;
        }
    }
}

// distance pass: TDM DMA (plane x) + async-to-LDS (plane y) staging, proposal bitmap,
// psum/usum via wave32 ballot+popcount
__global__ void k_dist(Scal* sc,
                       const float* __restrict__ embx, const float* __restrict__ emby,
                       const u32* __restrict__ m_bits, const u32* __restrict__ uncl_bits,
                       u32* __restrict__ prop_bits) {
    if (!sc->active || sc->stop) return;
    __shared__ float sEx[CCHUNK];
    __shared__ float sEy[CCHUNK];
    const float cx = sc->cx, cy = sc->cy, sx = sc->sx, sy = sc->sy;
    const int tid = threadIdx.x, lane = tid & 31;
    u32 accp = 0u, accu = 0u;
    const u32 ldsx = (u32)(uintptr_t)&sEx[0];
    const u32 ldsy = (u32)(uintptr_t)&sEy[0];
    for (int chunk = blockIdx.x * CCHUNK; chunk < NPIX; chunk += CBLOCKS * CCHUNK) {
        // plane y: per-thread async b128 (ASYNCcnt path), 512 x 16B slots
        for (int t = 0; t < 2; ++t) {
            int slot = tid + t * 256;
            async_b128(ldsy + slot * 16, emby + chunk + slot * 4);
        }
#if HAS_TDM
        // plane x: single TDM descriptor DMA (TENSORcnt path); EXEC-independent,
        // issued+drained by wave 0, published to the block by the barrier below.
        if (tid == 0) tdm_load_1d(ldsx, embx + chunk, CCHUNK);
#else
        for (int t = 0; t < 2; ++t) {
            int slot = tid + t * 256;
            async_b128(ldsx + slot * 16, embx + chunk + slot * 4);
        }
#endif
        wait_async0();
        __syncthreads();
        for (int sub = 0; sub < 8; ++sub) {
            int li = sub * 256 + tid;
            int p = chunk + li;
            float dx = sEx[li] - cx, dy = sEy[li] - cy;
            float q = dx * dx * sx + dy * dy * sy;
            u32 word = (u32)p >> 5;
            u32 mw = m_bits[word];
            bool prop = (q < LN2F) && ((mw >> lane) & 1u);
            u32 bal = ballot32(prop);
            if (lane == 0) {
                prop_bits[word] = bal;
                accp += (u32)__popc(bal);
                accu += (u32)__popc(bal & uncl_bits[word]);
            }
        }
        __syncthreads();   // safe to restage LDS
    }
    if (lane == 0 && (accp | accu)) {
        atomicAdd(&sc->psum, accp);
        atomicAdd(&sc->usum, accu);
    }
}

// accept/reject + apply mask/instance updates (one bitmap word per thread)
__global__ void k_apply(Scal* sc, const u32* __restrict__ prop_bits,
                        u32* __restrict__ uncl_bits, int* __restrict__ inst) {
    if (!sc->active || sc->stop) return;
    u32 psum = sc->psum, usum = sc->usum;
    int cs = sc->count_snapshot;
    u32 pd = psum > 1u ? psum : 1u;
    bool accept = (psum > MIN_INST_PIXEL) && ((float)usum / (float)pd > INST_RATIO);
    int w = blockIdx.x * 256 + threadIdx.x;   // grid covers NWORDS exactly
    if (w < NWORDS) {
        u32 pb = prop_bits[w];
        if (pb) {
            u32 uw = uncl_bits[w];
            u32 clear = uw & pb;
            if (clear) {                           // uncl2: proposal pixels leave pool
                uncl_bits[w] = uw & ~pb;
                atomicSub(&sc->uncl_count, (u32)__popc(clear));
            }
            if (accept) {                          // inst2 = count where take & proposal
                u32 b = pb; int base = w * 32;
                while (b) { int t = __ffs(b) - 1; inst[base + t] = cs; b &= b - 1u; }
            }
        }
    }
    if (blockIdx.x == 0 && threadIdx.x == 0 && accept) sc->count = cs + 1;
}

// uint8 cast then to harness float output
__global__ void k_final(const int* __restrict__ inst, float* __restrict__ out) {
    int i = blockIdx.x * 256 + threadIdx.x;
    out[i] = (float)(unsigned char)inst[i];
}

// ---------------- host ----------------
extern "C" void kernel_launch(void* const* d_in, const int* in_sizes, int n_in,
                              void* d_out, int out_size, void* d_ws, size_t ws_size,
                              hipStream_t stream) {
    const float* pred = (const float*)d_in[0];
    char* ws = (char*)d_ws;
    Scal* sc      = (Scal*)ws;
    float* partv  = (float*)(ws + 256);
    u32*   parti  = (u32*)(ws + 256 + 4096);
    size_t off = 16384;
    float* embx = (float*)(ws + off); off += (size_t)NPIX * 4;
    float* emby = (float*)(ws + off); off += (size_t)NPIX * 4;
    float* seed = (float*)(ws + off); off += (size_t)NPIX * 4;
    int*   inst = (int*)(ws + off);   off += (size_t)NPIX * 4;
    u32* m_bits    = (u32*)(ws + off); off += (size_t)NWORDS * 4;
    u32* uncl_bits = (u32*)(ws + off); off += (size_t)NWORDS * 4;
    u32* prop_bits = (u32*)(ws + off); off += (size_t)NWORDS * 4;
    float* out = (float*)d_out;

    k_init<<<1, 1, 0, stream>>>(sc);
    k_prep<<<NPIX / 256, 256, 0, stream>>>(pred, embx, emby, seed, inst, m_bits, uncl_bits, sc);
    for (int it = 0; it < MAX_ITER; ++it) {
        k_argmax<<<ABLOCKS, 256, 0, stream>>>(sc, seed, uncl_bits, partv, parti);
        k_select<<<1, 256, 0, stream>>>(sc, pred, embx, emby, partv, parti, uncl_bits);
        k_dist<<<CBLOCKS, 256, 0, stream>>>(sc, embx, emby, m_bits, uncl_bits, prop_bits);
        k_apply<<<NWORDS / 256, 256, 0, stream>>>(sc, prop_bits, uncl_bits, inst);
    }
    k_final<<<NPIX / 256, 256, 0, stream>>>(inst, out);
}